// UnifiedMonotonicSpline_61375082660149
// MI455X (gfx1250) — compile-verified
//
#include <hip/hip_runtime.h>
#include <stdint.h>

// Monotonic rational-quadratic spline eval (val, der) for MI455X / gfx1250.
// B=4096 rows, D=8192 points/row, W=129 weights/row, K=65 knots (64 segments).

#define W_DIM 129
#define D_DIM 8192
#define NSEG  32     // spline bins (2n)
#define NK    33     // base knots
#define KTOT  65     // interleaved knots
#define NREC  66     // 64 real segments + left/right virtual segments
#define LN2   0.6931471805599453f

typedef float v4f __attribute__((ext_vector_type(4)));
typedef float v2f __attribute__((ext_vector_type(2)));

// One spline evaluation. Top 3 search levels come from registers (row-uniform),
// next 2 levels probe the padded knot array in LDS, bottom 2 levels share one
// b128 probe, then two b128 record gathers. Extrapolation is folded into
// virtual records 0 and 65 with precomputed coefficients.
// Record layout: {xa, xb, wa*ya, wb*yb | wa, wb, wa*wb*(yb-ya)*(xb-xa), 0}.
// Returns {val, der}.
__device__ __forceinline__ v2f eval_one(
    float v,
    const float* __restrict__ sx, const float* __restrict__ rec,
    float t63, float t31, float t95,
    float t15, float t47, float t79, float t111)
{
    // j = count of knots <= v over the 128-padded sorted array
    const bool b1 = (t63 <= v);
    int j = b1 ? 64 : 0;
    const float l2 = b1 ? t95 : t31;
    const bool b2 = (l2 <= v);
    j += b2 ? 32 : 0;
    const float l3h = b1 ? t111 : t47;   // chosen when b2
    const float l3l = b1 ? t79  : t15;   // chosen when !b2
    const float l3  = b2 ? l3h : l3l;
    j += (l3 <= v) ? 16 : 0;
    j += (sx[j + 7] <= v) ? 8 : 0;
    j += (sx[j + 3] <= v) ? 4 : 0;
    // bottom two levels from one aligned b128 (j is a multiple of 4 here)
    const v4f q = *reinterpret_cast<const v4f*>(sx + j);
    const bool c2 = (q.y <= v);
    j += c2 ? 2 : 0;
    const float p1 = c2 ? q.z : q.x;
    j += (p1 <= v) ? 1 : 0;

    // record index: 0 = left virtual, 1..64 = real segs, 65 = right virtual
    const int m = j > (NREC - 1) ? (NREC - 1) : j;

    const v4f r0 = *reinterpret_cast<const v4f*>(rec + m * 8);      // xa,xb,A,B
    const v4f r1 = *reinterpret_cast<const v4f*>(rec + m * 8 + 4);  // wa,wb,C,_
    const float t0  = r0.y - v;
    const float t1  = v - r0.x;
    const float den = r1.x * t0 + r1.y * t1;
    const float rd  = __builtin_amdgcn_rcpf(den);

    v2f out;
    out.x = (r0.z * t0 + r0.w * t1) * rd;   // val
    out.y = r1.z * (rd * rd);               // der
    return out;
}

__global__ __launch_bounds__(256) void spline_eval_kernel(
    const float* __restrict__ in,   // (B, D)
    const float* __restrict__ wts,  // (B, W_DIM)
    float* __restrict__ out_val,    // (B, D)
    float* __restrict__ out_der)    // (B, D)
{
    const int r   = blockIdx.x;
    const int tid = threadIdx.x;

    __shared__ __align__(16) float sW[W_DIM + 3];      // raw row weights
    __shared__ float sBsx[NSEG], sBsy[NSEG], sLam[NSEG];
    __shared__ float sWb[NK];
    __shared__ float sCx[NK], sCy[NK];
    __shared__ __align__(16) float sx[128];            // knots x, padded to 128
    __shared__ float sy[KTOT + 1];
    __shared__ float swk[KTOT + 1];
    __shared__ __align__(16) float rec[NREC * 8];      // packed coefficients

    // ---- CDNA5 async copy: 129 weight floats -> LDS (ASYNCcnt-tracked) ----
    {
        const float* wrow = wts + (size_t)r * W_DIM;
        if (tid < W_DIM) {
            unsigned lds_off =
                (unsigned)(uintptr_t)(&sW[0]) + (unsigned)tid * 4u;
            unsigned long long ga =
                (unsigned long long)(uintptr_t)(wrow + tid);
            asm volatile("global_load_async_to_lds_b32 %0, %1, off"
                         :: "v"(lds_off), "v"(ga) : "memory");
        }
        asm volatile("s_wait_asynccnt 0" ::: "memory");
    }
    __syncthreads();

    // ---- stage 1: per-bin widths / lambdas / heights, base weights ----
    if (tid < NSEG) {
        const int s = tid;
        float wlo, whi, yv;
        if (s < 16) {                       // negative half: xn pairs, yn
            wlo = sW[32 + 2 * s];
            whi = sW[33 + 2 * s];
            yv  = sW[80 + s];
        } else {                            // positive half: xp pairs, yp
            const int p = s - 16;
            wlo = sW[2 * p];
            whi = sW[1 + 2 * p];
            yv  = sW[64 + p];
        }
        const float e0 = __expf(wlo - LN2);
        const float e1 = __expf(whi - LN2);
        const float bs = e0 + e1;
        sBsx[s] = bs;
        sLam[s] = e0 / fmaxf(bs, 1e-6f);
        sBsy[s] = __expf(yv);
    }
    if (tid >= 64 && tid < 64 + NK) {       // wb = exp(-0.5 * ln_d)
        const int k = tid - 64;
        sWb[k] = __expf(-0.5f * sW[96 + k]);
    }
    __syncthreads();

    // ---- stage 2: cumsums, centered at knot 16 (two lanes in parallel) ----
    if (tid == 0) {
        float acc = 0.f;
        sCx[0] = 0.f;
        for (int k = 0; k < NSEG; ++k) { acc += sBsx[k]; sCx[k + 1] = acc; }
        const float off = sCx[16];
        for (int k = 0; k < NK; ++k) sCx[k] -= off;
    } else if (tid == 32) {
        float acc = 0.f;
        sCy[0] = 0.f;
        for (int k = 0; k < NSEG; ++k) { acc += sBsy[k]; sCy[k + 1] = acc; }
        const float off = sCy[16];
        for (int k = 0; k < NK; ++k) sCy[k] -= off;
    }
    __syncthreads();

    // ---- stage 3: interleave base + mid knots into final arrays ----
    if (tid < NK) {
        sx[2 * tid]  = sCx[tid];
        sy[2 * tid]  = sCy[tid];
        swk[2 * tid] = sWb[tid];
    }
    if (tid >= 64 && tid < 64 + NSEG) {
        const int s = tid - 64;
        const float lam = sLam[s], bsx = sBsx[s], bsy = sBsy[s];
        const float wp = sWb[s], wn = sWb[s + 1];
        sx[2 * s + 1] = sCx[s] + lam * bsx;
        const float dm = (1.f - lam) * wp + lam * wn;
        sy[2 * s + 1] = sCy[s] + bsy * (lam * wn) / dm;
        const float numw = lam / wp + (1.f - lam) / wn;
        swk[2 * s + 1] = numw * bsx / fmaxf(bsy, 1e-6f);
    }
    if (tid >= 128 && tid < 128 + (128 - KTOT)) {   // pad sx[65..127] = +big
        sx[tid - 128 + KTOT] = 3.0e38f;
    }
    __syncthreads();

    // ---- stage 4: pack per-segment coefficient records (+ virtual segs) ----
    if (tid < 64) {
        const int s = tid;
        const float xa = sx[s], xb = sx[s + 1];
        const float ya = sy[s], yb = sy[s + 1];
        const float wa = swk[s], wb = swk[s + 1];
        v4f a; a.x = xa; a.y = xb; a.z = wa * ya; a.w = wb * yb;
        v4f b; b.x = wa; b.y = wb;
        b.z = (wa * wb) * (yb - ya) * (xb - xa); b.w = 0.f;
        *reinterpret_cast<v4f*>(rec + (s + 1) * 8)     = a;
        *reinterpret_cast<v4f*>(rec + (s + 1) * 8 + 4) = b;
    } else if (tid == 64) {
        // left virtual segment: unit weights, width 1 -> linear, slope d0
        const float x0 = sx[0], y0 = sy[0];
        const float d0 = __expf(sW[96]);
        v4f a; a.x = x0 - 1.f; a.y = x0; a.z = y0 - d0; a.w = y0;
        v4f b; b.x = 1.f; b.y = 1.f; b.z = d0; b.w = 0.f;
        *reinterpret_cast<v4f*>(rec + 0) = a;
        *reinterpret_cast<v4f*>(rec + 4) = b;
    } else if (tid == 65) {
        // right virtual segment: unit weights, width 1 -> linear, slope dN
        const float xN = sx[KTOT - 1], yN = sy[KTOT - 1];
        const float dN = __expf(sW[128]);
        v4f a; a.x = xN; a.y = xN + 1.f; a.z = yN; a.w = yN + dN;
        v4f b; b.x = 1.f; b.y = 1.f; b.z = dN; b.w = 0.f;
        *reinterpret_cast<v4f*>(rec + (NREC - 1) * 8)     = a;
        *reinterpret_cast<v4f*>(rec + (NREC - 1) * 8 + 4) = b;
    }
    __syncthreads();

    // ---- row-uniform top-of-tree knots held in registers ----
    const float t63 = sx[63], t31 = sx[31], t95 = sx[95];
    const float t15 = sx[15], t47 = sx[47], t79 = sx[79], t111 = sx[111];

    // ---- stream the 8192 points: NT float4 loads/stores + prefetch --------
    const size_t rowoff = (size_t)r * D_DIM;
    const v4f* __restrict__ vin  = reinterpret_cast<const v4f*>(in + rowoff);
    v4f* __restrict__ vval = reinterpret_cast<v4f*>(out_val + rowoff);
    v4f* __restrict__ vder = reinterpret_cast<v4f*>(out_der + rowoff);

    constexpr int CHUNKS = D_DIM / (256 * 4);   // 8
    #pragma unroll
    for (int c = 0; c < CHUNKS; ++c) {
        const int e = c * 256 + tid;
        const v4f v4 = __builtin_nontemporal_load(&vin[e]);
        if (c + 1 < CHUNKS) {
            __builtin_prefetch(&vin[e + 256], 0, 1);  // global_prefetch_b8
        }
        const v2f ra = eval_one(v4.x, sx, rec, t63, t31, t95, t15, t47, t79, t111);
        const v2f rb = eval_one(v4.y, sx, rec, t63, t31, t95, t15, t47, t79, t111);
        const v2f rc = eval_one(v4.z, sx, rec, t63, t31, t95, t15, t47, t79, t111);
        const v2f rd = eval_one(v4.w, sx, rec, t63, t31, t95, t15, t47, t79, t111);
        v4f val4, der4;
        val4.x = ra.x; val4.y = rb.x; val4.z = rc.x; val4.w = rd.x;
        der4.x = ra.y; der4.y = rb.y; der4.z = rc.y; der4.w = rd.y;
        __builtin_nontemporal_store(val4, &vval[e]);
        __builtin_nontemporal_store(der4, &vder[e]);
    }
}

extern "C" void kernel_launch(void* const* d_in, const int* in_sizes, int n_in,
                              void* d_out, int out_size, void* d_ws, size_t ws_size,
                              hipStream_t stream) {
    (void)n_in; (void)d_ws; (void)ws_size; (void)out_size;
    const float* in  = (const float*)d_in[0];   // (B, D) float32
    const float* wts = (const float*)d_in[1];   // (B, 129) float32

    const int B = in_sizes[1] / W_DIM;          // 4096
    const long long BD = (long long)B * D_DIM;

    float* out_val = (float*)d_out;             // first output (val)
    float* out_der = (float*)d_out + BD;        // second output (der)

    spline_eval_kernel<<<B, 256, 0, stream>>>(in, wts, out_val, out_der);
}